// DuplicationRemovalNetwork_8297876815894
// MI455X (gfx1250) — compile-verified
//
#include <hip/hip_runtime.h>
#include <math.h>

#define NROI 1024
#define NCLS 21
#define APPD 1024
#define DF   128
#define NH   16
#define DK   64

typedef float v2f __attribute__((ext_vector_type(2)));
typedef float v8f __attribute__((ext_vector_type(8)));

static __device__ __forceinline__ v8f wmma4(v2f a, v2f b, v8f c) {
  // D = A(16x4,f32) * B(4x16,f32) + C(16x16,f32)
  return __builtin_amdgcn_wmma_f32_16x16x4_f32(false, a, false, b, (short)0, c,
                                               false, false);
}

// A row-major [.,lda], B row-major [.,ldb]; accumulate 16x16 tile at (I,J).
static __device__ __forceinline__ void gemm_rr(const float* __restrict__ A, int lda,
                                               const float* __restrict__ B, int ldb,
                                               int I, int J, int K, v8f& c, int lane) {
  int ml = lane & 15, hi = lane >> 4;
  const float* arow = A + (size_t)(I + ml) * lda + 2 * hi;
  const float* bcol = B + J + ml;
  for (int kk = 0; kk < K; kk += 4) {
    v2f a = *(const v2f*)(arow + kk);
    v2f b;
    b.x = bcol[(size_t)(kk + 2 * hi) * ldb];
    b.y = bcol[(size_t)(kk + 2 * hi + 1) * ldb];
    c = wmma4(a, b, c);
  }
}

// A row-major, B given transposed (Bt[n][k], contiguous in k) -> A * Bt^T
static __device__ __forceinline__ void gemm_rt(const float* __restrict__ A, int lda,
                                               const float* __restrict__ Bt, int ldb,
                                               int I, int J, int K, v8f& c, int lane) {
  int ml = lane & 15, hi = lane >> 4;
  const float* arow = A + (size_t)(I + ml) * lda + 2 * hi;
  const float* brow = Bt + (size_t)(J + ml) * ldb + 2 * hi;
  for (int kk = 0; kk < K; kk += 4) {
    v2f a = *(const v2f*)(arow + kk);
    v2f b = *(const v2f*)(brow + kk);
    c = wmma4(a, b, c);
  }
}

static __device__ __forceinline__ void store_tile(float* __restrict__ C, int ldc,
                                                  int I, int J, v8f c, int lane) {
  int nl = lane & 15, hi = lane >> 4;
#pragma unroll
  for (int r = 0; r < 8; ++r)
    C[(size_t)(I + r + 8 * hi) * ldc + J + nl] = c[r];
}

// ---------------- stage 1: decode + softmax/argmax ----------------
__global__ void k_decode(const float* __restrict__ roi,
                         const float* __restrict__ cls_loc,
                         const float* __restrict__ score,
                         const int* __restrict__ size,
                         float* __restrict__ prob, int* __restrict__ label,
                         float* __restrict__ bbox) {
  int i = blockIdx.x * blockDim.x + threadIdx.x;
  if (i >= NROI) return;
  const float* s = score + i * NCLS;
  float m = s[0];
  int c = 0;
  for (int j = 1; j < NCLS; ++j) {
    float v = s[j];
    if (v > m) { m = v; c = j; }
  }
  float sum = 0.f;
  for (int j = 0; j < NCLS; ++j) sum += __expf(s[j] - m);
  prob[i] = 1.0f / sum;   // max of softmax
  label[i] = c;
  float y0 = roi[i * 4 + 0], x0 = roi[i * 4 + 1];
  float y1 = roi[i * 4 + 2], x1 = roi[i * 4 + 3];
  float h = y1 - y0, w = x1 - x0;
  float cy = y0 + 0.5f * h, cx = x0 + 0.5f * w;
  const float* l = cls_loc + i * NCLS * 4 + c * 4;
  float dy = l[0] * 0.1f, dx = l[1] * 0.1f, dh = l[2] * 0.2f, dw = l[3] * 0.2f;
  float ncy = dy * h + cy, ncx = dx * w + cx;
  float nh = __expf(dh) * h, nw = __expf(dw) * w;
  float b0 = ncy - 0.5f * nh, b1 = ncx - 0.5f * nw;
  float b2 = ncy + 0.5f * nh, b3 = ncx + 0.5f * nw;
  // reference clamps class-strided slices of [N,21,4]: even class -> size[0]
  float bound = (float)size[(c & 1) ? 1 : 0];
  b0 = fminf(fmaxf(b0, 0.f), bound);
  b1 = fminf(fmaxf(b1, 0.f), bound);
  b2 = fminf(fmaxf(b2, 0.f), bound);
  b3 = fminf(fmaxf(b3, 0.f), bound);
  bbox[i * 4 + 0] = b0; bbox[i * 4 + 1] = b1;
  bbox[i * 4 + 2] = b2; bbox[i * 4 + 3] = b3;
}

// ---------------- stage 2: bitonic sort (descending prob, stable) --------
__global__ void k_sort(const float* __restrict__ prob, int* __restrict__ order,
                       float* __restrict__ sprob) {
  __shared__ unsigned long long keys[NROI];
  int t = threadIdx.x;
  unsigned u = __float_as_uint(prob[t]);
  u = (u & 0x80000000u) ? ~u : (u | 0x80000000u);   // monotone-ascending map
  keys[t] = (((unsigned long long)(~u)) << 32) | (unsigned)t;  // asc => prob desc
  __syncthreads();
  for (int k = 2; k <= NROI; k <<= 1) {
    for (int j = k >> 1; j > 0; j >>= 1) {
      int ixj = t ^ j;
      if (ixj > t) {
        unsigned long long a = keys[t], b = keys[ixj];
        bool up = ((t & k) == 0);
        if ((a > b) == up) { keys[t] = b; keys[ixj] = a; }
      }
      __syncthreads();
    }
  }
  int o = (int)(keys[t] & 0xffffffffu);
  order[t] = o;
  sprob[t] = prob[o];
}

// ---------------- stage 3: gather + emit labels/boxes --------------------
__global__ void k_gather(const float* __restrict__ feat, const int* __restrict__ order,
                         const int* __restrict__ label, const float* __restrict__ bbox,
                         float* __restrict__ sfeat, float* __restrict__ sbbox,
                         float* __restrict__ out) {
  int i = blockIdx.x, t = threadIdx.x;
  int o = order[i];
  for (int d = t; d < APPD; d += blockDim.x)
    sfeat[(size_t)i * APPD + d] = feat[(size_t)o * APPD + d];
  if (t < 4) {
    float b = bbox[o * 4 + t];
    sbbox[i * 4 + t] = b;
    out[2 * NROI + i * 4 + t] = b;   // output #3
  }
  if (t == 0) out[NROI + i] = (float)(label[o] - 1);  // output #2
}

// ---------------- stage 4: rank embedding (constant sin/cos) -------------
__global__ void k_rank(float* __restrict__ rank) {
  int i = blockIdx.x, t = threadIdx.x;
  for (int d = t; d < APPD; d += blockDim.x) {
    int r = d & 511;
    float dm = __expf(-(float)r * (6.90775528f / 512.0f));  // 1000^(-r/512)
    float mul = (float)i * dm;
    rank[(size_t)i * APPD + d] = (d < 512) ? __sinf(mul) : __cosf(mul);
  }
}

// ---------------- stage 5: emb = sfeat@feat_w + rank@rank_w + biases -----
__global__ void __launch_bounds__(32)
k_embfeat(const float* __restrict__ sfeat, const float* __restrict__ rank,
          const float* __restrict__ feat_w, const float* __restrict__ feat_b,
          const float* __restrict__ rank_w, const float* __restrict__ rank_b,
          float* __restrict__ emb) {
  int lane = threadIdx.x;
  int I = blockIdx.x * 16, J = blockIdx.y * 16;
  int nl = lane & 15;
  v8f c;
  float b = feat_b[J + nl] + rank_b[J + nl];
#pragma unroll
  for (int r = 0; r < 8; ++r) c[r] = b;
  gemm_rr(sfeat, APPD, feat_w, DF, I, J, APPD, c, lane);
  gemm_rr(rank, APPD, rank_w, DF, I, J, APPD, c, lane);
  store_tile(emb, DF, I, J, c, lane);
}

// ---------------- stage 6: per-head K/Q/V projections --------------------
__global__ void __launch_bounds__(32)
k_headproj(const float* __restrict__ emb, const float* __restrict__ W,
           const float* __restrict__ bias, float* __restrict__ outm) {
  int lane = threadIdx.x;
  int h = blockIdx.z;
  int I = blockIdx.x * 16, J = blockIdx.y * 16;
  int nl = lane & 15;
  v8f c;
  float b = bias[h * DK + J + nl];
#pragma unroll
  for (int r = 0; r < 8; ++r) c[r] = b;
  gemm_rr(emb, DF, W + (size_t)h * DF * DK, DK, I, J, DF, c, lane);
  store_tile(outm + (size_t)h * NROI * DK, DK, I, J, c, lane);
}

// ---------------- stage 7: fused pos-emb + geo heads -> log(w_g) ---------
__global__ void k_wg(const float* __restrict__ sbbox, const float* __restrict__ wg_w,
                     const float* __restrict__ wg_b, float* __restrict__ L) {
  __shared__ float sw[NH * 64];
  __shared__ float sb[NH];
  int t = threadIdx.x;
  for (int j = t; j < NH * 64; j += 256) sw[j] = wg_w[j];
  if (t < NH) sb[t] = wg_b[t];
  __syncthreads();
  int m = blockIdx.y;
  int n = blockIdx.x * 256 + t;
  float bm0 = sbbox[m * 4 + 0], bm1 = sbbox[m * 4 + 1];
  float bm2 = sbbox[m * 4 + 2], bm3 = sbbox[m * 4 + 3];
  float bn0 = sbbox[n * 4 + 0], bn1 = sbbox[n * 4 + 1];
  float bn2 = sbbox[n * 4 + 2], bn3 = sbbox[n * 4 + 3];
  float cxm = (bm0 + bm2) * 0.5f, cym = (bm1 + bm3) * 0.5f;
  float wm = bm2 - bm0 + 1.0f, hm = bm3 - bm1 + 1.0f;
  float cxn = (bn0 + bn2) * 0.5f, cyn = (bn1 + bn3) * 0.5f;
  float wn = bn2 - bn0 + 1.0f, hn = bn3 - bn1 + 1.0f;
  float pos[4];
  pos[0] = __logf(fmaxf(fabsf((cxm - cxn) / wm), 1e-3f));
  pos[1] = __logf(fmaxf(fabsf((cym - cyn) / hm), 1e-3f));
  pos[2] = __logf(wm / wn);
  pos[3] = __logf(hm / hn);
  float acc[NH];
#pragma unroll
  for (int h = 0; h < NH; ++h) acc[h] = sb[h];
#pragma unroll
  for (int p = 0; p < 4; ++p) {
    float base = 100.0f * pos[p];
#pragma unroll
    for (int r = 0; r < 8; ++r) {
      float dm = __expf(-(float)r * 0.86346941f);  // 1000^(-r/8)
      float mul = base * dm;
      float sv = __sinf(mul), cv = __cosf(mul);
      int j = p * 8 + r;
#pragma unroll
      for (int h = 0; h < NH; ++h)
        acc[h] += sv * sw[h * 64 + j] + cv * sw[h * 64 + 32 + j];
    }
  }
  size_t off = ((size_t)m << 10) + (size_t)n;
#pragma unroll
  for (int h = 0; h < NH; ++h) {
    float wv = fmaxf(acc[h], 0.0f);                  // relu
    L[((size_t)h << 20) + off] = __logf(fmaxf(wv, 1e-6f));
  }
}

// ---------------- stage 8: L += (K . Q^T) / sqrt(64) via WMMA ------------
__global__ void __launch_bounds__(32)
k_scores(const float* __restrict__ Km, const float* __restrict__ Qm,
         float* __restrict__ L) {
  int lane = threadIdx.x;
  int h = blockIdx.z;
  int I = blockIdx.x * 16, J = blockIdx.y * 16;
  v8f c;
#pragma unroll
  for (int r = 0; r < 8; ++r) c[r] = 0.0f;
  gemm_rt(Km + ((size_t)h << 16), DK, Qm + ((size_t)h << 16), DK, I, J, DK, c, lane);
  int nl = lane & 15, hi = lane >> 4;
#pragma unroll
  for (int r = 0; r < 8; ++r) {
    size_t idx = ((size_t)h << 20) + (size_t)(I + r + 8 * hi) * NROI + J + nl;
    L[idx] += c[r] * 0.125f;
  }
}

// ---------------- stage 9: in-place row softmax over n -------------------
__global__ void k_softmax(float* __restrict__ L) {
  __shared__ float red[256];
  int t = threadIdx.x;
  size_t base = (size_t)blockIdx.x << 10;
  float v0 = L[base + t], v1 = L[base + t + 256];
  float v2 = L[base + t + 512], v3 = L[base + t + 768];
  float lm = fmaxf(fmaxf(v0, v1), fmaxf(v2, v3));
  red[t] = lm;
  __syncthreads();
  for (int s = 128; s > 0; s >>= 1) {
    if (t < s) red[t] = fmaxf(red[t], red[t + s]);
    __syncthreads();
  }
  float m = red[0];
  __syncthreads();
  v0 = __expf(v0 - m); v1 = __expf(v1 - m);
  v2 = __expf(v2 - m); v3 = __expf(v3 - m);
  red[t] = v0 + v1 + v2 + v3;
  __syncthreads();
  for (int s = 128; s > 0; s >>= 1) {
    if (t < s) red[t] += red[t + s];
    __syncthreads();
  }
  float inv = 1.0f / red[0];
  L[base + t] = v0 * inv;       L[base + t + 256] = v1 * inv;
  L[base + t + 512] = v2 * inv; L[base + t + 768] = v3 * inv;
}

// ---------------- stage 10: rel = P @ V via WMMA -------------------------
__global__ void __launch_bounds__(32)
k_rel(const float* __restrict__ L, const float* __restrict__ Vm,
      float* __restrict__ relm) {
  int lane = threadIdx.x;
  int h = blockIdx.z;
  int I = blockIdx.x * 16, J = blockIdx.y * 16;
  v8f c;
#pragma unroll
  for (int r = 0; r < 8; ++r) c[r] = 0.0f;
  gemm_rr(L + ((size_t)h << 20), NROI, Vm + ((size_t)h << 16), DK, I, J, NROI, c, lane);
  store_tile(relm + ((size_t)h << 16), DK, I, J, c, lane);
}

// ---------------- stage 11: sigmoid gate * sorted_prob -------------------
__global__ void __launch_bounds__(32)
k_final(const float* __restrict__ relm, const float* __restrict__ sfeat,
        const float* __restrict__ logit_w, const float* __restrict__ logit_b,
        const float* __restrict__ sprob, float* __restrict__ out) {
  int i = blockIdx.x, lane = threadIdx.x;
  float acc = 0.f;
  for (int f = lane; f < NH * DK; f += 32) {
    int h = f >> 6, kd = f & 63;
    float v = relm[((size_t)h << 16) + (size_t)i * DK + kd] +
              sfeat[(size_t)i * APPD + f];
    acc += v * logit_w[f];
  }
  for (int off = 16; off > 0; off >>= 1) acc += __shfl_down(acc, off);
  if (lane == 0) {
    float s1 = 1.0f / (1.0f + __expf(-(acc + logit_b[0])));
    out[i] = s1 * sprob[i];   // output #1
  }
}

extern "C" void kernel_launch(void* const* d_in, const int* in_sizes, int n_in,
                              void* d_out, int out_size, void* d_ws, size_t ws_size,
                              hipStream_t stream) {
  (void)in_sizes; (void)n_in; (void)out_size; (void)ws_size;
  const float* sample_roi  = (const float*)d_in[0];
  const float* roi_cls_loc = (const float*)d_in[1];
  const float* roi_score   = (const float*)d_in[2];
  const float* app_feat    = (const float*)d_in[3];
  const int*   size        = (const int*)d_in[4];
  const float* rank_w      = (const float*)d_in[5];
  const float* rank_b      = (const float*)d_in[6];
  const float* feat_w      = (const float*)d_in[7];
  const float* feat_b      = (const float*)d_in[8];
  const float* logit_w     = (const float*)d_in[9];
  const float* logit_b     = (const float*)d_in[10];
  const float* wg_w        = (const float*)d_in[11];
  const float* wg_b        = (const float*)d_in[12];
  const float* wk_w        = (const float*)d_in[13];
  const float* wk_b        = (const float*)d_in[14];
  const float* wq_w        = (const float*)d_in[15];
  const float* wq_b        = (const float*)d_in[16];
  const float* wv_w        = (const float*)d_in[17];
  const float* wv_b        = (const float*)d_in[18];
  float* out = (float*)d_out;

  float* ws = (float*)d_ws;
  float* prob     = ws;                       // 1024 f
  int*   label    = (int*)(ws + 1024);        // 1024 i
  int*   order    = (int*)(ws + 2048);        // 1024 i
  float* sprob    = ws + 3072;                // 1024 f
  float* bbox_all = ws + 4096;                // 4096 f
  float* sbbox    = ws + 8192;                // 4096 f
  float* sfeat    = ws + 16384;               // 1M f
  float* rankm    = sfeat + (1 << 20);        // 1M f
  float* emb      = rankm + (1 << 20);        // 128K f
  float* Km       = emb + (NROI * DF);        // 1M f
  float* Qm       = Km + (1 << 20);           // 1M f
  float* Vm       = Qm + (1 << 20);           // 1M f
  float* relm     = Vm + (1 << 20);           // 1M f
  float* L        = relm + (1 << 20);         // 16M f (16x1024x1024)

  k_decode<<<NROI / 256, 256, 0, stream>>>(sample_roi, roi_cls_loc, roi_score, size,
                                           prob, label, bbox_all);
  k_sort<<<1, NROI, 0, stream>>>(prob, order, sprob);
  k_gather<<<NROI, 256, 0, stream>>>(app_feat, order, label, bbox_all, sfeat, sbbox, out);
  k_rank<<<NROI, 256, 0, stream>>>(rankm);
  k_embfeat<<<dim3(NROI / 16, DF / 16), 32, 0, stream>>>(sfeat, rankm, feat_w, feat_b,
                                                         rank_w, rank_b, emb);
  k_headproj<<<dim3(NROI / 16, DK / 16, NH), 32, 0, stream>>>(emb, wk_w, wk_b, Km);
  k_headproj<<<dim3(NROI / 16, DK / 16, NH), 32, 0, stream>>>(emb, wq_w, wq_b, Qm);
  k_headproj<<<dim3(NROI / 16, DK / 16, NH), 32, 0, stream>>>(emb, wv_w, wv_b, Vm);
  k_wg<<<dim3(NROI / 256, NROI), 256, 0, stream>>>(sbbox, wg_w, wg_b, L);
  k_scores<<<dim3(NROI / 16, NROI / 16, NH), 32, 0, stream>>>(Km, Qm, L);
  k_softmax<<<NH * NROI, 256, 0, stream>>>(L);
  k_rel<<<dim3(NROI / 16, DK / 16, NH), 32, 0, stream>>>(L, Vm, relm);
  k_final<<<NROI, 32, 0, stream>>>(relm, sfeat, logit_w, logit_b, sprob, out);
}